// NeighborhoodAttentionS2_14010183319908
// MI455X (gfx1250) — compile-verified
//
#include <hip/hip_runtime.h>
#include <hip/hip_bf16.h>
#include <math.h>

#define NLAT 46
#define NLON 90
#define CH   256
#define NPOS (NLAT * NLON)   // 4140

typedef __attribute__((ext_vector_type(2))) float v2f;
typedef __attribute__((ext_vector_type(8))) float v8f;

#define SLAB_STRIDE 20   // floats per K-row in LDS: 16 data + 4 pad
                         // -> 80 B row stride: keeps async b128 writes 16B-aligned
                         // and makes the 16-lane column reads bank-conflict-free.

// -----------------------------------------------------------------------------
// Kernel 1: fused Q/K/V projection GEMM with bias, V_WMMA_F32_16X16X4_F32.
//   OUT[which](pos, ch) = sum_c W[ch, c] * X[c, pos] + bias[ch]
// Block = 8 waves. One block owns ONE 16-wide N (position) tile and 8 of the
// 16 M (channel) tiles; the shared B slab (256 K x 16 cols, 16 KB) is staged
// into LDS once via async global->LDS b128 transfers and reused by all waves.
// No wave-divergent control flow anywhere -> EXEC stays all-1s (WMMA req).
// -----------------------------------------------------------------------------
__global__ __launch_bounds__(256) void qkv_proj_wmma(
    const float* __restrict__ qo, const float* __restrict__ ki,
    const float* __restrict__ vi,
    const float* __restrict__ qw, const float* __restrict__ kw,
    const float* __restrict__ vw,
    const float* __restrict__ qb, const float* __restrict__ kb,
    const float* __restrict__ vb,
    float* __restrict__ ws)
{
    const int which = blockIdx.y;                 // 0=q, 1=k, 2=v
    const float* __restrict__ X = (which == 0) ? qo : (which == 1) ? ki : vi;
    const float* __restrict__ W = (which == 0) ? qw : (which == 1) ? kw : vw;
    const float* __restrict__ B = (which == 0) ? qb : (which == 1) ? kb : vb;
    float* __restrict__ OUT = ws + (size_t)which * CH * NPOS;   // (pos, ch)

    __shared__ float slab[CH * SLAB_STRIDE];      // 20480 B

    const int nt    = blockIdx.x >> 1;            // N tile (0..258)
    const int mhalf = blockIdx.x & 1;             // which 8 M tiles
    const int n0    = nt * 16;

    // ---- Stage B slab: X[k][n0..n0+15] for k = 0..255, async -> LDS ----
    // 1024 x b128 transfers (4 per thread). Edge tile: clamp the *global*
    // address into bounds; the garbage columns feed only unstored D columns.
    {
        const long gmax = (long)CH * NPOS - 4;
#pragma unroll
        for (int i = 0; i < 4; ++i) {
            const int idx = threadIdx.x + i * 256;  // 0..1023
            const int row = idx >> 2;               // K index 0..255
            const int seg = idx & 3;                // 4-float segment 0..3
            long goff = (long)row * NPOS + n0 + seg * 4;
            if (goff > gmax) goff = gmax;           // stays 16B-aligned
            const float* gptr = X + goff;
            const unsigned laddr =
                (unsigned)(size_t)&slab[row * SLAB_STRIDE + seg * 4];
            asm volatile("global_load_async_to_lds_b128 %0, %1, off"
                         :: "v"(laddr), "v"(gptr) : "memory");
        }
        asm volatile("s_wait_asynccnt 0" ::: "memory");
        __syncthreads();
    }

    const int lane   = threadIdx.x & 31;
    const int wave   = threadIdx.x >> 5;
    const int mt     = mhalf * 8 + wave;          // M tile for this wave
    const int m0     = mt * 16;
    const int hiHalf = lane >> 4;                 // 0: K pair {0,1}, 1: {2,3}
    const int l15    = lane & 15;
    const int mrow   = m0 + l15;                  // A row this lane supplies
    const int ncol   = n0 + l15;                  // B/D column of this lane

    const float* __restrict__ Arow = W + (size_t)mrow * CH;

    v8f acc = {};
    for (int k0 = 0; k0 < CH; k0 += 4) {
        const int kk = k0 + hiHalf * 2;
        // A (16x4 f32): lanes 0-15 -> K {k0,k0+1}, lanes 16-31 -> {k0+2,k0+3}
        v2f a;
        a.x = Arow[kk];
        a.y = Arow[kk + 1];
        // B (4x16 f32) from LDS, mirrored K split, N = lane&15
        v2f b;
        b.x = slab[kk * SLAB_STRIDE + l15];
        b.y = slab[(kk + 1) * SLAB_STRIDE + l15];
        acc = __builtin_amdgcn_wmma_f32_16x16x4_f32(
            /*neg_a=*/false, a, /*neg_b=*/false, b,
            /*c_mod=*/(short)0, acc, /*reuse_a=*/false, /*reuse_b=*/false);
    }

    if (ncol < NPOS) {
        // C/D layout: VGPR r holds M = m0 + r + (lane>=16 ? 8 : 0), N = lane&15
#pragma unroll
        for (int r = 0; r < 8; ++r) {
            const int mm = m0 + r + hiHalf * 8;
            OUT[(size_t)ncol * CH + mm] = acc[r] + B[mm];
        }
    }
}

// -----------------------------------------------------------------------------
// Kernel 2: sparse neighborhood attention with online softmax.
// One 256-thread block per output point (t, w); one channel per thread.
// -----------------------------------------------------------------------------
__global__ __launch_bounds__(256) void attn_s2_kernel(
    const float* __restrict__ ws,
    const float* __restrict__ quad_w,
    const int*   __restrict__ row_ids,
    const int*   __restrict__ col_idx,
    int nnz,
    float* __restrict__ out)
{
    const float* __restrict__ qbuf = ws;                          // (pos, ch)
    const float* __restrict__ kbuf = ws + (size_t)CH * NPOS;
    const float* __restrict__ vbuf = ws + 2 * (size_t)CH * NPOS;

    const int w = blockIdx.x;          // 0..NLON-1
    const int t = blockIdx.y;          // 0..NLAT-1
    const int c = threadIdx.x;         // 0..CH-1
    const int lane = c & 31;
    const int wv   = c >> 5;

    __shared__ float red[8];

    // Binary search [start, end) of row t in sorted row_ids.
    int lo = 0, hi = nnz;
    while (lo < hi) { int mid = (lo + hi) >> 1; if (row_ids[mid] <  t) lo = mid + 1; else hi = mid; }
    const int start = lo;
    lo = start; hi = nnz;
    while (lo < hi) { int mid = (lo + hi) >> 1; if (row_ids[mid] <= t) lo = mid + 1; else hi = mid; }
    const int end = lo;

    const int p = t * NLON + w;
    const float qv = qbuf[(size_t)p * CH + c];

    float mrun = -INFINITY;   // running max of scores
    float drun = 0.0f;        // running weighted denominator
    float acc  = 0.0f;        // running weighted numerator (this channel)

    for (int n = start; n < end; ++n) {
        const int ci = col_idx[n];
        const int h  = ci / NLON;
        int wi = (ci % NLON) + w;
        if (wi >= NLON) wi -= NLON;
        const int pk = h * NLON + wi;

        // score = sum_c q[c] * k[c]  (wave shuffle + 8-slot LDS combine)
        float prod = qv * kbuf[(size_t)pk * CH + c];
#pragma unroll
        for (int off = 16; off > 0; off >>= 1)
            prod += __shfl_xor(prod, off, 32);
        if (lane == 0) red[wv] = prod;
        __syncthreads();
        const float s = red[0] + red[1] + red[2] + red[3] +
                        red[4] + red[5] + red[6] + red[7];
        __syncthreads();

        // Online softmax update, weighted by source-latitude quadrature weight.
        const float vv   = vbuf[(size_t)pk * CH + c];
        const float quad = quad_w[h];
        const float nm   = fmaxf(mrun, s);
        const float so   = __expf(mrun - nm);      // 0 when mrun == -inf
        const float e    = __expf(s - nm) * quad;
        drun = drun * so + e;
        acc  = acc  * so + e * vv;
        mrun = nm;
    }

    // out layout: (b=1, ch, nlat, nlon)
    out[(size_t)c * NPOS + p] = acc / drun;
}

// -----------------------------------------------------------------------------
// Launch
//   inputs: 0 qo, 1 ki, 2 vi, 3 q_w, 4 k_w, 5 v_w, 6 q_b, 7 k_b, 8 v_b,
//           9 quad_w, 10 row_ids(int32), 11 col_idx(int32)
//   workspace: 3 * 256 * 4140 * 4 B ~= 12.1 MB (q/k/v in (pos, ch) layout)
// -----------------------------------------------------------------------------
extern "C" void kernel_launch(void* const* d_in, const int* in_sizes, int n_in,
                              void* d_out, int out_size, void* d_ws, size_t ws_size,
                              hipStream_t stream)
{
    const float* qo  = (const float*)d_in[0];
    const float* ki  = (const float*)d_in[1];
    const float* vi  = (const float*)d_in[2];
    const float* q_w = (const float*)d_in[3];
    const float* k_w = (const float*)d_in[4];
    const float* v_w = (const float*)d_in[5];
    const float* q_b = (const float*)d_in[6];
    const float* k_b = (const float*)d_in[7];
    const float* v_b = (const float*)d_in[8];
    const float* quad    = (const float*)d_in[9];
    const int*   row_ids = (const int*)d_in[10];
    const int*   col_idx = (const int*)d_in[11];
    const int    nnz     = in_sizes[10];

    float* ws = (float*)d_ws;

    const int NT = (NPOS + 15) / 16;        // 259 N tiles
    dim3 gProj(NT * 2, 3);                  // x: (N tile, M half), y: q/k/v
    qkv_proj_wmma<<<gProj, 256, 0, stream>>>(qo, ki, vi, q_w, k_w, v_w,
                                             q_b, k_b, v_b, ws);

    dim3 gAttn(NLON, NLAT);
    attn_s2_kernel<<<gAttn, 256, 0, stream>>>(ws, quad, row_ids, col_idx,
                                              nnz, (float*)d_out);
}